// PerfectSpatialHash_84164179133378
// MI455X (gfx1250) — compile-verified
//
#include <hip/hip_runtime.h>
#include <hip/hip_bf16.h>

// Perfect spatial hash gather:
//   oidx = trunc(coords * m1) mod 64   -> offsets = offset_table[oidx]   (3 MB, cache-resident)
//   hidx = (trunc(coords * m0) + offsets) mod 256
//   out  = hash_table[hidx]            (512 MB, random 32B gathers -> HBM/L2 bound)
//
// Strategy: 4 queries/thread for memory-level parallelism; non-temporal
// (TH=NT, CDNA5 ISA 7.3) loads for streamed coords and NT stores for the
// streamed output so the 192MB L2 is reserved for the 512MB hash table.
// Power-of-two floored modulo == bitwise AND (matches jnp % semantics,
// including negative operands).

#define HASH_SIZE 256
#define OFFSET_SIZE 64
#define FEAT 8
#define QPT 4  // queries per thread

// Native clang vectors: required by __builtin_nontemporal_load/store
// (HIP_vector_type structs are rejected by the builtin).
typedef int   vint4   __attribute__((ext_vector_type(4)));
typedef float vfloat4 __attribute__((ext_vector_type(4)));

__global__ __launch_bounds__(256) void PerfectSpatialHash_84164179133378_kernel(
    const int* __restrict__ coords,        // (N,3) int32
    const float* __restrict__ hash_table,  // (256,256,256,8) f32
    const int* __restrict__ offset_table,  // (64,64,64,3) int32
    const float* __restrict__ m0p,         // (3,) f32
    const float* __restrict__ m1p,         // (3,) f32
    float* __restrict__ out,               // (N,8) f32
    int n)
{
    const float m0x = m0p[0], m0y = m0p[1], m0z = m0p[2];
    const float m1x = m1p[0], m1y = m1p[1], m1z = m1p[2];

    const int t    = blockIdx.x * blockDim.x + threadIdx.x;
    const int base = t * QPT;
    if (base >= n) return;

    int cx[QPT], cy[QPT], cz[QPT];

    if (base + QPT <= n) {
        // 4 queries = 48B of coords = 3 x b128, 16B-aligned. Non-temporal:
        // streamed once, keep it out of L2.
        const vint4* cp = reinterpret_cast<const vint4*>(coords + (size_t)base * 3);
        vint4 a = __builtin_nontemporal_load(cp + 0);
        vint4 b = __builtin_nontemporal_load(cp + 1);
        vint4 c = __builtin_nontemporal_load(cp + 2);
        cx[0] = a.x; cy[0] = a.y; cz[0] = a.z;
        cx[1] = a.w; cy[1] = b.x; cz[1] = b.y;
        cx[2] = b.z; cy[2] = b.w; cz[2] = c.x;
        cx[3] = c.y; cy[3] = c.z; cz[3] = c.w;
    } else {
        #pragma unroll
        for (int q = 0; q < QPT; ++q) {
            const int i = base + q;
            if (i < n) {
                cx[q] = coords[3 * (size_t)i + 0];
                cy[q] = coords[3 * (size_t)i + 1];
                cz[q] = coords[3 * (size_t)i + 2];
            } else {
                cx[q] = 0; cy[q] = 0; cz[q] = 0;
            }
        }
    }

    // --- Stage 1: offset-table gathers (12 independent b32 loads in flight).
    // Table is 3MB -> lives in L2/WGP$; regular temporal hint on purpose.
    int offx[QPT], offy[QPT], offz[QPT];
    #pragma unroll
    for (int q = 0; q < QPT; ++q) {
        // trunc-toward-zero int cast matches jnp .astype(int32);
        // floored %64 on int32 == &63 (power of two).
        const int ox = ((int)((float)cx[q] * m1x)) & (OFFSET_SIZE - 1);
        const int oy = ((int)((float)cy[q] * m1y)) & (OFFSET_SIZE - 1);
        const int oz = ((int)((float)cz[q] * m1z)) & (OFFSET_SIZE - 1);
        const unsigned oi = ((((unsigned)ox << 6) | (unsigned)oy) << 6) | (unsigned)oz;
        const int* op = offset_table + (size_t)oi * 3;
        offx[q] = op[0];
        offy[q] = op[1];
        offz[q] = op[2];
    }

    // --- Stage 2: hash-table gathers (8 independent b128 loads in flight).
    // 512MB table vs 192MB L2: these are the dominant traffic; default RT
    // temporal hint so hits stick in L2.
    vfloat4 f0[QPT], f1[QPT];
    #pragma unroll
    for (int q = 0; q < QPT; ++q) {
        const int hx = (((int)((float)cx[q] * m0x)) + offx[q]) & (HASH_SIZE - 1);
        const int hy = (((int)((float)cy[q] * m0y)) + offy[q]) & (HASH_SIZE - 1);
        const int hz = (((int)((float)cz[q] * m0z)) + offz[q]) & (HASH_SIZE - 1);
        const unsigned hi = ((unsigned)hx << 16) | ((unsigned)hy << 8) | (unsigned)hz;
        const vfloat4* hp = reinterpret_cast<const vfloat4*>(hash_table + (size_t)hi * FEAT);
        f0[q] = hp[0];
        f1[q] = hp[1];
    }

    // --- Stage 3: streamed output, non-temporal stores (don't evict the
    // hash table's L2 working set). 128B contiguous per thread.
    if (base + QPT <= n) {
        vfloat4* dst = reinterpret_cast<vfloat4*>(out + (size_t)base * FEAT);
        #pragma unroll
        for (int q = 0; q < QPT; ++q) {
            __builtin_nontemporal_store(f0[q], dst + 2 * q + 0);
            __builtin_nontemporal_store(f1[q], dst + 2 * q + 1);
        }
    } else {
        #pragma unroll
        for (int q = 0; q < QPT; ++q) {
            const int i = base + q;
            if (i < n) {
                vfloat4* dst = reinterpret_cast<vfloat4*>(out + (size_t)i * FEAT);
                __builtin_nontemporal_store(f0[q], dst + 0);
                __builtin_nontemporal_store(f1[q], dst + 1);
            }
        }
    }
}

extern "C" void kernel_launch(void* const* d_in, const int* in_sizes, int n_in,
                              void* d_out, int out_size, void* d_ws, size_t ws_size,
                              hipStream_t stream) {
    const int*   coords       = (const int*)d_in[0];
    const float* hash_table   = (const float*)d_in[1];
    const int*   offset_table = (const int*)d_in[2];
    const float* m0           = (const float*)d_in[3];
    const float* m1           = (const float*)d_in[4];
    float*       out          = (float*)d_out;

    const int n = in_sizes[0] / 3;  // N_QUERIES
    const int threads = (n + QPT - 1) / QPT;
    const int block = 256;
    const int grid = (threads + block - 1) / block;

    PerfectSpatialHash_84164179133378_kernel<<<grid, block, 0, stream>>>(
        coords, hash_table, offset_table, m0, m1, out, n);
}